// MoE_82952998355167
// MI455X (gfx1250) — compile-verified
//
#include <hip/hip_runtime.h>
#include <hip/hip_bf16.h>

typedef float v2f __attribute__((ext_vector_type(2)));
typedef float v8f __attribute__((ext_vector_type(8)));

// ---------------- problem sizes ----------------
#define S_TOK 4096
#define MDIM 512
#define NE 16
#define TOPK 2
#define CDIM 128
#define HH1 500
#define HH2 500
#define HH3 2000
#define NS 3685            // SAMPLE
#define NTOT (2*NS)        // 7370 rows in MMD "total"
#define NPAD 7424          // = 116*64, zero-padded rows
#define NT (NPAD/16)       // 464 i-tiles
#define JGRP (NPAD/64)     // 116 j-groups (4 tiles each)

// ---------------- workspace layout (float offsets), ~3.98 MB total ----------------
#define OFF_GATE 0                 // S*2 floats
#define OFF_IDX 8192               // S*2 ints
#define OFF_SELSUM 16384           // 16
#define OFF_HITS 16400             // 16
#define OFF_SCAL 16416             // [0]=sumsq [1]=csum2 [2]=distacc [3..7]=bwinv[5]
#define OFF_EIN 16448              // E*2*M = 16384
#define OFF_EOUT 32832             // E*2*128 = 4096
#define OFF_SQ 36928               // NPAD
#define OFF_TOTAL 44352            // NPAD*128 = 950272

// ---------------- output layout ----------------
#define OUT_OUT 0
#define OUT_SEL0 (S_TOK*CDIM)                  // 524288
#define OUT_BAL (OUT_SEL0 + S_TOK*NE*TOPK)    // 655360
#define OUT_DIST (OUT_BAL + 1)

__global__ void k_zero(float* ws) {
    int t = threadIdx.x;
    if (t < 16) { ws[OFF_SELSUM + t] = 0.f; ws[OFF_HITS + t] = 0.f; }
    if (t < 8)  ws[OFF_SCAL + t] = 0.f;
}

// router: select = x@Wr.T + noise ; top-2 ; dispatch output ; balance accumulators
__global__ void k_router(const float* __restrict__ x, const float* __restrict__ noise,
                         const float* __restrict__ Wr, float* __restrict__ ws,
                         float* __restrict__ out) {
    __shared__ float wr[NE * MDIM];  // 32 KB
    for (int i = threadIdx.x; i < NE * MDIM; i += blockDim.x) wr[i] = Wr[i];
    __syncthreads();
    int s = blockIdx.x * blockDim.x + threadIdx.x;
    float sel[NE];
#pragma unroll
    for (int e = 0; e < NE; e++) sel[e] = noise[s * NE + e];
    const float* xr = x + (size_t)s * MDIM;
    for (int m = 0; m < MDIM; m++) {
        float xv = xr[m];
#pragma unroll
        for (int e = 0; e < NE; e++) sel[e] = fmaf(xv, wr[e * MDIM + m], sel[e]);
    }
    // top-2, ties -> lower index first (lax.top_k semantics)
    int i0 = 0; float g0 = sel[0];
#pragma unroll
    for (int e = 1; e < NE; e++) if (sel[e] > g0) { g0 = sel[e]; i0 = e; }
    int i1 = (i0 == 0) ? 1 : 0; float g1 = sel[i1];
#pragma unroll
    for (int e = 0; e < NE; e++) if (e != i0 && sel[e] > g1) { g1 = sel[e]; i1 = e; }
    ws[OFF_GATE + s * 2 + 0] = g0;
    ws[OFF_GATE + s * 2 + 1] = g1;
    int* idx = (int*)(ws + OFF_IDX);
    idx[s * 2 + 0] = i0; idx[s * 2 + 1] = i1;
    // select0 = dispatch = (combine != 0), shape (S,E,K)
#pragma unroll
    for (int e = 0; e < NE; e++) {
        out[OUT_SEL0 + (size_t)s * NE * TOPK + e * TOPK + 0] = (e == i0 && g0 != 0.f) ? 1.f : 0.f;
        out[OUT_SEL0 + (size_t)s * NE * TOPK + e * TOPK + 1] = (e == i1 && g1 != 0.f) ? 1.f : 0.f;
    }
#pragma unroll
    for (int e = 0; e < NE; e++) atomicAdd(&ws[OFF_SELSUM + e], sel[e]);
    atomicAdd(&ws[OFF_HITS + i0], 1.f);
    atomicAdd(&ws[OFF_HITS + i1], 1.f);
}

// expert_inputs[e,k,m] = sum_s dispatch[s,e,k] * x[s,m]
__global__ void k_dispatch(const float* __restrict__ x, float* __restrict__ ws) {
    int e = blockIdx.x >> 1;
    int k = blockIdx.x & 1;
    int t = threadIdx.x;  // 0..511 = m
    const float* gate = ws + OFF_GATE;
    const int* idx = (const int*)(ws + OFF_IDX);
    float acc = 0.f;
    for (int s = 0; s < S_TOK; s++) {
        if (idx[s * 2 + k] == e && gate[s * 2 + k] != 0.f) acc += x[(size_t)s * MDIM + t];
    }
    ws[OFF_EIN + (e * 2 + k) * MDIM + t] = acc;
}

// 4-layer MLP per expert, both K slots at once (weight-row reuse)
__global__ void k_mlp(const float* __restrict__ W1, const float* __restrict__ b1,
                      const float* __restrict__ W2, const float* __restrict__ b2,
                      const float* __restrict__ W3, const float* __restrict__ b3,
                      const float* __restrict__ W4, const float* __restrict__ b4,
                      float* __restrict__ ws) {
    __shared__ float xin[2][MDIM];
    __shared__ float sh1[2][HH1];
    __shared__ float sh2[2][HH2];
    __shared__ float sh3[2][HH3];
    int e = blockIdx.x;
    int t = threadIdx.x;  // 0..511
    xin[0][t] = ws[OFF_EIN + (e * 2 + 0) * MDIM + t];
    xin[1][t] = ws[OFF_EIN + (e * 2 + 1) * MDIM + t];
    __syncthreads();
    if (t < HH1) {
        const float* w = W1 + ((size_t)e * HH1 + t) * MDIM;
        float a0 = b1[e * HH1 + t], a1 = a0;
        for (int m = 0; m < MDIM; m++) { float wv = w[m]; a0 = fmaf(wv, xin[0][m], a0); a1 = fmaf(wv, xin[1][m], a1); }
        sh1[0][t] = fmaxf(a0, 0.f); sh1[1][t] = fmaxf(a1, 0.f);
    }
    __syncthreads();
    if (t < HH2) {
        const float* w = W2 + ((size_t)e * HH2 + t) * HH1;
        float a0 = b2[e * HH2 + t], a1 = a0;
        for (int m = 0; m < HH1; m++) { float wv = w[m]; a0 = fmaf(wv, sh1[0][m], a0); a1 = fmaf(wv, sh1[1][m], a1); }
        sh2[0][t] = fmaxf(a0, 0.f); sh2[1][t] = fmaxf(a1, 0.f);
    }
    __syncthreads();
    for (int f = t; f < HH3; f += 512) {
        const float* w = W3 + ((size_t)e * HH3 + f) * HH2;
        float a0 = b3[e * HH3 + f], a1 = a0;
        for (int g = 0; g < HH2; g++) { float wv = w[g]; a0 = fmaf(wv, sh2[0][g], a0); a1 = fmaf(wv, sh2[1][g], a1); }
        sh3[0][f] = fmaxf(a0, 0.f); sh3[1][f] = fmaxf(a1, 0.f);
    }
    __syncthreads();
    if (t < CDIM) {
        const float* w = W4 + ((size_t)e * CDIM + t) * HH3;
        float a0 = b4[e * CDIM + t], a1 = a0;
        for (int f = 0; f < HH3; f++) { float wv = w[f]; a0 = fmaf(wv, sh3[0][f], a0); a1 = fmaf(wv, sh3[1][f], a1); }
        ws[OFF_EOUT + (e * 2 + 0) * CDIM + t] = a0;
        ws[OFF_EOUT + (e * 2 + 1) * CDIM + t] = a1;
    }
}

// output[s,:] = sum_k gate[s,k]*eout[idx[s,k],k,:]
__global__ void k_output(float* __restrict__ out, const float* __restrict__ ws) {
    int s = blockIdx.x, t = threadIdx.x;
    const float* gate = ws + OFF_GATE;
    const int* idx = (const int*)(ws + OFF_IDX);
    float g0 = gate[s * 2], g1 = gate[s * 2 + 1];
    int i0 = idx[s * 2], i1 = idx[s * 2 + 1];
    out[OUT_OUT + (size_t)s * CDIM + t] =
        g0 * ws[OFF_EOUT + (i0 * 2 + 0) * CDIM + t] + g1 * ws[OFF_EOUT + (i1 * 2 + 1) * CDIM + t];
}

// total = concat(middle[0][idx1], middle[1][idx2]); zero-pad to NPAD rows
__global__ void k_gather(const int* __restrict__ idx1, const int* __restrict__ idx2,
                         float* __restrict__ ws) {
    int r = blockIdx.x, t = threadIdx.x;
    const int* idx = (const int*)(ws + OFF_IDX);
    float v = 0.f;
    if (r < NS) {
        int s = idx1[r];
        float g = ws[OFF_GATE + s * 2 + 0];
        int e = idx[s * 2 + 0];
        v = g * ws[OFF_EOUT + (e * 2 + 0) * CDIM + t];
    } else if (r < NTOT) {
        int s = idx2[r - NS];
        float g = ws[OFF_GATE + s * 2 + 1];
        int e = idx[s * 2 + 1];
        v = g * ws[OFF_EOUT + (e * 2 + 1) * CDIM + t];
    }
    ws[OFF_TOTAL + (size_t)r * CDIM + t] = v;
}

__global__ void k_sq(float* __restrict__ ws) {
    int i = blockIdx.x * blockDim.x + threadIdx.x;
    if (i >= NPAD) return;
    const float* row = ws + OFF_TOTAL + (size_t)i * CDIM;
    float s = 0.f;
    for (int m = 0; m < CDIM; m++) s = fmaf(row[m], row[m], s);
    ws[OFF_SQ + i] = s;
    if (i < NTOT) atomicAdd(&ws[OFF_SCAL + 0], s);
}

__global__ void k_colsum(float* __restrict__ ws) {
    __shared__ float red[CDIM];
    int m = threadIdx.x;
    float c = 0.f;
    for (int i = 0; i < NTOT; i++) c += ws[OFF_TOTAL + (size_t)i * CDIM + m];
    red[m] = c * c;
    __syncthreads();
    for (int off = 64; off > 0; off >>= 1) { if (m < off) red[m] += red[m + off]; __syncthreads(); }
    if (m == 0) ws[OFF_SCAL + 1] = red[0];
}

// bw from closed form: sum(d2) = 2n*sum(sq) - 2*sum_m(colsum^2); also balance loss
__global__ void k_scalars(float* __restrict__ ws, float* __restrict__ out) {
    float sumsq = ws[OFF_SCAL + 0], csum2 = ws[OFF_SCAL + 1];
    float n = (float)NTOT;
    float sum_d2 = 2.f * n * sumsq - 2.f * csum2;
    float bw = sum_d2 / (n * n - n);
    bw *= 0.25f;  // / KMUL^(KNUM//2) = / 2^2
#pragma unroll
    for (int t = 0; t < 5; t++) ws[OFF_SCAL + 3 + t] = 1.f / (bw * (float)(1 << t));
    float bal = 0.f;
#pragma unroll
    for (int e = 0; e < NE; e++) bal += (ws[OFF_SELSUM + e] / (float)S_TOK) * ws[OFF_HITS + e];
    out[OUT_BAL] = bal * (float)NE / ((float)S_TOK * (float)TOPK);
}

// Fused MMD pass: WMMA f32 Gram tiles -> d2 -> 5-kernel exp -> signed sum.
// One wave computes a 16x64 strip (1 A-tile x 4 B-tiles).
__global__ __launch_bounds__(256) void k_mmd(const float* __restrict__ total,
                                             const float* __restrict__ sq,
                                             const float* __restrict__ scal,
                                             float* __restrict__ distacc) {
    int wave = blockIdx.x * (blockDim.x >> 5) + (threadIdx.x >> 5);
    if (wave >= NT * JGRP) return;                 // wave-uniform; EXEC all-1s below
    int it = wave / JGRP, jg = wave - it * JGRP;
    int lane = threadIdx.x & 31;
    int row = lane & 15;
    int hi = lane >> 4;                            // K-half select
    int i0 = it * 16, j0 = jg * 64;
    const float* arow = total + (size_t)(i0 + row) * CDIM + 2 * hi;
    const float* brow = total + (size_t)(j0 + row) * CDIM + 2 * hi;
    v8f acc[4] = { {0,0,0,0,0,0,0,0}, {0,0,0,0,0,0,0,0}, {0,0,0,0,0,0,0,0}, {0,0,0,0,0,0,0,0} };
    for (int kk = 0; kk < CDIM; kk += 4) {
        v2f a = *(const v2f*)(arow + kk);
#pragma unroll
        for (int jt = 0; jt < 4; jt++) {
            v2f b = *(const v2f*)(brow + (size_t)jt * 16 * CDIM + kk);
            acc[jt] = __builtin_amdgcn_wmma_f32_16x16x4_f32(
                false, a, false, b, (short)0, acc[jt], false, false);
        }
    }
    float bw0 = scal[3], bw1 = scal[4], bw2 = scal[5], bw3 = scal[6], bw4 = scal[7];
    int ibase = i0 + 8 * hi;                       // C/D layout: M = r + 8*hi
    float sqi[8];
#pragma unroll
    for (int r = 0; r < 8; r++) sqi[r] = sq[ibase + r];
    float local = 0.f;
#pragma unroll
    for (int jt = 0; jt < 4; jt++) {
        int j = j0 + jt * 16 + row;
        float sqj = sq[j];
        float jin = (j < NTOT) ? 1.f : 0.f;
        float jsrc = (j < NS) ? 1.f : 0.f;
#pragma unroll
        for (int r = 0; r < 8; r++) {
            int i = ibase + r;
            float d2 = sqi[r] + sqj - 2.f * acc[jt][r];
            float ksum = __expf(-d2 * bw0) + __expf(-d2 * bw1) + __expf(-d2 * bw2) +
                         __expf(-d2 * bw3) + __expf(-d2 * bw4);
            float iin = (i < NTOT) ? 1.f : 0.f;
            float isrc = (i < NS) ? 1.f : 0.f;
            float sign = (isrc == jsrc) ? 1.f : -1.f;
            local += iin * jin * sign * ksum;
        }
    }
#pragma unroll
    for (int off = 16; off > 0; off >>= 1) local += __shfl_down(local, off, 32);
    if (lane == 0) atomicAdd(distacc, local);
}

__global__ void k_final(const float* __restrict__ ws, float* __restrict__ out) {
    out[OUT_DIST] = -ws[OFF_SCAL + 2] / ((float)NS * (float)NS);
}

extern "C" void kernel_launch(void* const* d_in, const int* in_sizes, int n_in,
                              void* d_out, int out_size, void* d_ws, size_t ws_size,
                              hipStream_t stream) {
    const float* x    = (const float*)d_in[0];
    const float* noise= (const float*)d_in[1];
    const float* Wr   = (const float*)d_in[2];
    const float* W1   = (const float*)d_in[3];
    const float* b1   = (const float*)d_in[4];
    const float* W2   = (const float*)d_in[5];
    const float* b2   = (const float*)d_in[6];
    const float* W3   = (const float*)d_in[7];
    const float* b3   = (const float*)d_in[8];
    const float* W4   = (const float*)d_in[9];
    const float* b4   = (const float*)d_in[10];
    const int* idx1   = (const int*)d_in[11];
    const int* idx2   = (const int*)d_in[12];
    float* out = (float*)d_out;
    float* ws  = (float*)d_ws;   // needs ~4 MB

    k_zero<<<1, 32, 0, stream>>>(ws);
    k_router<<<S_TOK / 256, 256, 0, stream>>>(x, noise, Wr, ws, out);
    k_dispatch<<<NE * TOPK, MDIM, 0, stream>>>(x, ws);
    k_mlp<<<NE, 512, 0, stream>>>(W1, b1, W2, b2, W3, b3, W4, b4, ws);
    k_output<<<S_TOK, CDIM, 0, stream>>>(out, ws);
    k_gather<<<NPAD, CDIM, 0, stream>>>(idx1, idx2, ws);
    k_sq<<<(NPAD + 255) / 256, 256, 0, stream>>>(ws);
    k_colsum<<<1, CDIM, 0, stream>>>(ws);
    k_scalars<<<1, 1, 0, stream>>>(ws, out);
    int waves = NT * JGRP;
    k_mmd<<<(waves + 7) / 8, 256, 0, stream>>>(ws + OFF_TOTAL, ws + OFF_SQ,
                                               ws + OFF_SCAL, ws + OFF_SCAL + 2);
    k_final<<<1, 1, 0, stream>>>(ws, out);
}